// GAT_32719060861013
// MI455X (gfx1250) — compile-verified
//
#include <hip/hip_runtime.h>
#include <cstdint>
#include <cstddef>

// GAT on MI455X (gfx1250): f32 WMMA GEMM + bandwidth-bound edge passes.
// N=100000 nodes, D_IN = H*C = 128, H=2 heads of C=64.

typedef __attribute__((ext_vector_type(2))) float v2f;
typedef __attribute__((ext_vector_type(8))) float v8f;

// ---------------------------------------------------------------------------
// GEMM: Hout[M x 128] = X[M x 128] @ W[128 x 128], exact f32 via
// V_WMMA_F32_16X16X4_F32. One wave computes a 16x16 tile; 8 waves per block
// cover all 128 output columns of a 16-row stripe. M must be a multiple of 16
// (100000 = 6250*16).
// ---------------------------------------------------------------------------
__global__ __launch_bounds__(256) void gat_gemm128(
    const float* __restrict__ X, const float* __restrict__ W,
    float* __restrict__ Hout, int M) {
  __shared__ float xt[16 * 128];  // 8 KB x-tile
  const int m0 = blockIdx.x * 16;
  const int t = threadIdx.x;

  // Cooperative coalesced load of the 16x128 x tile (512 float4s, 256 thr).
  {
    const float4* src = (const float4*)(X + (size_t)m0 * 128);
    float4* dst = (float4*)xt;
    dst[t] = src[t];
    dst[t + 256] = src[t + 256];
  }
  __syncthreads();

  const int wave = t >> 5;          // 0..7 -> output column tile
  const int lane = t & 31;
  const int f0 = wave * 16;
  const int half = lane >> 4;       // lanes 16-31 hold K+2,K+3 slices
  const int l15 = lane & 15;

  v8f acc = {};
  for (int k = 0; k < 128; k += 4) {
    v2f a, b;
    // A 16x4 (MxK): lane holds (M=l15, K=k+half*2) and (M=l15, K=k+half*2+1)
    a.x = xt[l15 * 128 + k + half * 2 + 0];
    a.y = xt[l15 * 128 + k + half * 2 + 1];
    // B 4x16 (KxN): row striped across lanes within a VGPR
    b.x = W[(k + half * 2 + 0) * 128 + f0 + l15];
    b.y = W[(k + half * 2 + 1) * 128 + f0 + l15];
    acc = __builtin_amdgcn_wmma_f32_16x16x4_f32(
        /*neg_a=*/false, a, /*neg_b=*/false, b,
        /*c_mod=*/(short)0, acc, /*reuse_a=*/false, /*reuse_b=*/false);
  }

  // C/D layout: VGPR r, lanes 0-15 -> row r; lanes 16-31 -> row r+8.
#pragma unroll
  for (int r = 0; r < 8; ++r) {
    Hout[(size_t)(m0 + r + half * 8) * 128 + f0 + l15] = acc[r];
  }
}

// ---------------------------------------------------------------------------
// Per-node attention scalars: asrc[n,h] = sum_c h[n,h,c]*a_src[h,c] (and adst).
// One wave32 per node; a_src/a_dst flat (H*C = 128, feature-major) so
// feature f maps directly to aS[f]/aD[f].
// ---------------------------------------------------------------------------
__global__ __launch_bounds__(256) void gat_node_scores(
    const float* __restrict__ Hn, const float* __restrict__ aS,
    const float* __restrict__ aD, float* __restrict__ asrc,
    float* __restrict__ adst, int n) {
  const int wave = threadIdx.x >> 5, lane = threadIdx.x & 31;
  const int node = blockIdx.x * 8 + wave;
  if (node >= n) return;
  const float* hp = Hn + (size_t)node * 128;
  const float h0a = hp[lane], h0b = hp[lane + 32];
  const float h1a = hp[lane + 64], h1b = hp[lane + 96];
  float s0 = h0a * aS[lane] + h0b * aS[lane + 32];
  float s1 = h1a * aS[lane + 64] + h1b * aS[lane + 96];
  float d0 = h0a * aD[lane] + h0b * aD[lane + 32];
  float d1 = h1a * aD[lane + 64] + h1b * aD[lane + 96];
#pragma unroll
  for (int off = 16; off; off >>= 1) {
    s0 += __shfl_down(s0, off);
    s1 += __shfl_down(s1, off);
    d0 += __shfl_down(d0, off);
    d1 += __shfl_down(d1, off);
  }
  if (lane == 0) {
    asrc[node * 2 + 0] = s0;
    asrc[node * 2 + 1] = s1;
    adst[node * 2 + 0] = d0;
    adst[node * 2 + 1] = d1;
  }
}

// ---------------------------------------------------------------------------
// Monotonic float<->uint encoding so segment-max can use atomicMax(u32).
// ---------------------------------------------------------------------------
__device__ __forceinline__ uint32_t fenc(float f) {
  uint32_t b = __float_as_uint(f);
  return (b & 0x80000000u) ? ~b : (b | 0x80000000u);
}
__device__ __forceinline__ float fdec(uint32_t u) {
  return __uint_as_float((u & 0x80000000u) ? (u & 0x7FFFFFFFu) : ~u);
}

__global__ void gat_fill_u32(uint32_t* __restrict__ p, uint32_t v, size_t n) {
  size_t i = (size_t)blockIdx.x * blockDim.x + threadIdx.x;
  if (i < n) p[i] = v;
}

// Pass 1: e = leaky_relu(asrc[src]+adst[dst], 0.2); store e; segment max (dst).
__global__ void gat_edge_max(const int* __restrict__ srcI,
                             const int* __restrict__ dstI,
                             const float* __restrict__ asrc,
                             const float* __restrict__ adst,
                             float* __restrict__ ebuf,
                             uint32_t* __restrict__ mEnc, int E, int Etot) {
  int tid = blockIdx.x * blockDim.x + threadIdx.x;
  int ge = tid >> 1, hh = tid & 1;
  if (ge >= Etot) return;
  int s = (ge < E) ? srcI[ge] : (ge - E);  // appended self-loops
  int d = (ge < E) ? dstI[ge] : (ge - E);
  float e = asrc[s * 2 + hh] + adst[d * 2 + hh];
  e = e > 0.0f ? e : 0.2f * e;
  ebuf[(size_t)ge * 2 + hh] = e;
  atomicMax(&mEnc[d * 2 + hh], fenc(e));
}

// Pass 2: p = exp(e - m[dst]); store p in-place; segment sum (dst).
__global__ void gat_edge_exp(const int* __restrict__ dstI,
                             float* __restrict__ ebuf,
                             const uint32_t* __restrict__ mEnc,
                             float* __restrict__ sbuf, int E, int Etot) {
  int tid = blockIdx.x * blockDim.x + threadIdx.x;
  int ge = tid >> 1, hh = tid & 1;
  if (ge >= Etot) return;
  int d = (ge < E) ? dstI[ge] : (ge - E);
  float m = fdec(mEnc[d * 2 + hh]);
  float p = __expf(ebuf[(size_t)ge * 2 + hh] - m);
  ebuf[(size_t)ge * 2 + hh] = p;
  atomicAdd(&sbuf[d * 2 + hh], p);
}

// Pass 3: acc[dst] += h[src] * alpha. One wave per edge; lane l owns features
// 4l..4l+3 (lane<16 -> head 0, lane>=16 -> head 1). Coalesced 512B gather.
__global__ __launch_bounds__(256) void gat_edge_scatter(
    const int* __restrict__ srcI, const int* __restrict__ dstI,
    const float* __restrict__ Hn, const float* __restrict__ ebuf,
    const float* __restrict__ sbuf, float* __restrict__ acc, int E, int Etot) {
  const int wave = threadIdx.x >> 5, lane = threadIdx.x & 31;
  const int ge = blockIdx.x * 8 + wave;
  if (ge >= Etot) return;
  int s = (ge < E) ? srcI[ge] : (ge - E);
  int d = (ge < E) ? dstI[ge] : (ge - E);
  const int hh = lane >> 4;
  const float alpha =
      ebuf[(size_t)ge * 2 + hh] / (sbuf[d * 2 + hh] + 1e-16f);
  const float4 hv = *(const float4*)(Hn + (size_t)s * 128 + lane * 4);
  float* ap = acc + (size_t)d * 128 + lane * 4;
  atomicAdd(ap + 0, hv.x * alpha);
  atomicAdd(ap + 1, hv.y * alpha);
  atomicAdd(ap + 2, hv.z * alpha);
  atomicAdd(ap + 3, hv.w * alpha);
}

// Epilogue: acc = relu(acc + bias), in place.
__global__ void gat_bias_relu(float* __restrict__ acc,
                              const float* __restrict__ b, size_t total) {
  size_t i = (size_t)blockIdx.x * blockDim.x + threadIdx.x;
  if (i < total) {
    float v = acc[i] + b[i & 127];
    acc[i] = v > 0.0f ? v : 0.0f;
  }
}

// ---------------------------------------------------------------------------
// Driver
// ---------------------------------------------------------------------------
extern "C" void kernel_launch(void* const* d_in, const int* in_sizes, int n_in,
                              void* d_out, int out_size, void* d_ws,
                              size_t ws_size, hipStream_t stream) {
  const float* x = (const float*)d_in[0];
  const int* ei = (const int*)d_in[1];   // [2,E] flat
  const float* W1 = (const float*)d_in[2];
  const float* aS1 = (const float*)d_in[3];
  const float* aD1 = (const float*)d_in[4];
  const float* b1 = (const float*)d_in[5];
  const float* W2 = (const float*)d_in[6];
  const float* aS2 = (const float*)d_in[7];
  const float* aD2 = (const float*)d_in[8];
  const float* b2 = (const float*)d_in[9];
  float* out = (float*)d_out;

  const int N = in_sizes[0] / 128;
  const int E = in_sizes[1] / 2;
  const int Etot = E + N;  // appended self-loops
  const int* srcI = ei;
  const int* dstI = ei + E;

  // Workspace bump allocator (256B aligned).
  size_t off = 0;
  auto alloc = [&](size_t bytes) -> void* {
    void* p = (char*)d_ws + off;
    off += (bytes + 255) & ~(size_t)255;
    return p;
  };
  float* hbuf = (float*)alloc((size_t)N * 128 * 4);      // x@W (pre-attn)
  float* acc1 = (float*)alloc((size_t)N * 128 * 4);      // layer-1 out
  float* ebuf = (float*)alloc((size_t)Etot * 2 * 4);     // e then p, per head
  float* asrc = (float*)alloc((size_t)N * 2 * 4);
  float* adst = (float*)alloc((size_t)N * 2 * 4);
  uint32_t* mEnc = (uint32_t*)alloc((size_t)N * 2 * 4);
  float* sbuf = (float*)alloc((size_t)N * 2 * 4);
  (void)ws_size;

  const uint32_t NEG_INF_ENC = 0x007FFFFFu;  // fenc(-inf)
  const size_t nodeF = (size_t)N * 128;

  auto run_layer = [&](const float* Xin, const float* W, const float* aS,
                       const float* aD, const float* bias, float* accOut) {
    gat_gemm128<<<N / 16, 256, 0, stream>>>(Xin, W, hbuf, N);
    gat_node_scores<<<(N + 7) / 8, 256, 0, stream>>>(hbuf, aS, aD, asrc, adst,
                                                     N);
    gat_fill_u32<<<(int)((N * 2 + 255) / 256), 256, 0, stream>>>(
        mEnc, NEG_INF_ENC, (size_t)N * 2);
    gat_fill_u32<<<(int)((N * 2 + 255) / 256), 256, 0, stream>>>(
        (uint32_t*)sbuf, 0u, (size_t)N * 2);
    gat_fill_u32<<<(int)((nodeF + 255) / 256), 256, 0, stream>>>(
        (uint32_t*)accOut, 0u, nodeF);
    gat_edge_max<<<(Etot * 2 + 255) / 256, 256, 0, stream>>>(
        srcI, dstI, asrc, adst, ebuf, mEnc, E, Etot);
    gat_edge_exp<<<(Etot * 2 + 255) / 256, 256, 0, stream>>>(dstI, ebuf, mEnc,
                                                             sbuf, E, Etot);
    gat_edge_scatter<<<(Etot + 7) / 8, 256, 0, stream>>>(srcI, dstI, hbuf, ebuf,
                                                         sbuf, accOut, E, Etot);
    gat_bias_relu<<<(int)((nodeF + 255) / 256), 256, 0, stream>>>(accOut, bias,
                                                                  nodeF);
  };

  run_layer(x, W1, aS1, aD1, b1, acc1);    // layer 1 -> acc1
  run_layer(acc1, W2, aS2, aD2, b2, out);  // layer 2 -> d_out
}